// SphericalFourierLayer_53815940218923
// MI455X (gfx1250) — compile-verified
//
#include <hip/hip_runtime.h>
#include <cmath>

typedef float v2f __attribute__((ext_vector_type(2)));
typedef float v8f __attribute__((ext_vector_type(8)));

#define B_     4
#define H_     128
#define MODES_ 256
#define NC_    65536

__device__ __forceinline__ float gelu_exact(float v) {
    return 0.5f * v * (1.0f + erff(v * 0.70710678118654752440f));
}

// ---------------------------------------------------------------------------
// Kernel 1: spectral per-mode complex matmul for n < MODES.
//   Yr[k, b*256+m] = sum_h x[b,h,m] * w_real[h,k,m] + bias[k,m]
//   Yi[k, b*256+m] = sum_h x[b,h,m] * w_imag[h,k,m]
// lane <-> mode offset (coalesced w_* reads); x tile staged in LDS.
// grid: MODES/32 blocks of 256 threads (8 waves, each owns 16 k's).
// ---------------------------------------------------------------------------
__global__ void __launch_bounds__(256)
sfno_spectral_kernel(const float* __restrict__ x,
                     const float* __restrict__ w_real,
                     const float* __restrict__ w_imag,
                     const float* __restrict__ bias,
                     float* __restrict__ Yr,
                     float* __restrict__ Yi)
{
    __shared__ float xs[H_ * 32 * B_];      // xs[(h*32+mo)*4 + b], 64 KB
    const int tid = threadIdx.x;
    const int m0  = blockIdx.x * 32;

    // cooperative coalesced load of x[:, :, m0:m0+32]
    #pragma unroll 4
    for (int i = 0; i < 64; ++i) {
        int e  = i * 256 + tid;             // e = ((b*128 + h)*32 + mo)
        int mo = e & 31;
        int h  = (e >> 5) & (H_ - 1);
        int b  = e >> 12;
        xs[(h * 32 + mo) * 4 + b] = x[(size_t)(b * H_ + h) * NC_ + m0 + mo];
    }
    __syncthreads();

    const int lane = tid & 31;
    const int wave = tid >> 5;
    const int m    = m0 + lane;

    for (int k = wave * 16; k < wave * 16 + 16; ++k) {
        const float bb = bias[(size_t)k * NC_ + m];
        float aR0 = bb, aR1 = bb, aR2 = bb, aR3 = bb;
        float aI0 = 0.f, aI1 = 0.f, aI2 = 0.f, aI3 = 0.f;
        #pragma unroll 4
        for (int h = 0; h < H_; ++h) {
            const float wr = w_real[(size_t)(h * H_ + k) * MODES_ + m];
            const float wi = w_imag[(size_t)(h * H_ + k) * MODES_ + m];
            const float4 xv = *reinterpret_cast<const float4*>(&xs[(h * 32 + lane) * 4]);
            aR0 = fmaf(xv.x, wr, aR0);  aI0 = fmaf(xv.x, wi, aI0);
            aR1 = fmaf(xv.y, wr, aR1);  aI1 = fmaf(xv.y, wi, aI1);
            aR2 = fmaf(xv.z, wr, aR2);  aI2 = fmaf(xv.z, wi, aI2);
            aR3 = fmaf(xv.w, wr, aR3);  aI3 = fmaf(xv.w, wi, aI3);
        }
        const int row = k * (B_ * MODES_);
        Yr[row + 0 * MODES_ + m] = aR0;  Yi[row + 0 * MODES_ + m] = aI0;
        Yr[row + 1 * MODES_ + m] = aR1;  Yi[row + 1 * MODES_ + m] = aI1;
        Yr[row + 2 * MODES_ + m] = aR2;  Yi[row + 2 * MODES_ + m] = aI2;
        Yr[row + 3 * MODES_ + m] = aR3;  Yi[row + 3 * MODES_ + m] = aI3;
    }
}

// ---------------------------------------------------------------------------
// Kernel 2: head matmul for n < MODES using fp32 WMMA.
//   Z[k2, (b,n)] = gelu(sum_k lin_w[k2,k] * Y[k,(b,n)] (+ lin_b for real))
// One wave = one 16x16 tile; dual accumulators (real & imag) share A frags.
// grid: (B*MODES)/16 = 64 blocks of 256 threads (8 waves = 8 k2-tiles).
// ---------------------------------------------------------------------------
__global__ void __launch_bounds__(256)
sfno_head_kernel(const float* __restrict__ Yr,
                 const float* __restrict__ Yi,
                 const float* __restrict__ lin_w,
                 const float* __restrict__ lin_b,
                 float* __restrict__ out)
{
    const int lane   = threadIdx.x & 31;
    const int wave   = threadIdx.x >> 5;
    const int laneLo = lane & 15;
    const int laneHi = lane >> 4;
    const int kb     = wave << 4;                 // k2 tile base
    const int bm0    = blockIdx.x << 4;           // (b,n) column tile base

    v8f accR = {};
    v8f accI = {};
    for (int k = 0; k < H_; k += 4) {
        v2f a, br, bi;
        // A = lin_w[k2 = kb+laneLo][k + 2*laneHi + {0,1}]  (16x4, M x K)
        const float* ap = lin_w + (kb + laneLo) * H_ + k + 2 * laneHi;
        a.x = ap[0];  a.y = ap[1];
        // B = Y[k + 2*laneHi + {0,1}][bm0 + laneLo]        (4x16, K x N)
        const float* rp = Yr + (k + 2 * laneHi) * (B_ * MODES_) + bm0 + laneLo;
        br.x = rp[0]; br.y = rp[B_ * MODES_];
        const float* ip = Yi + (k + 2 * laneHi) * (B_ * MODES_) + bm0 + laneLo;
        bi.x = ip[0]; bi.y = ip[B_ * MODES_];
        accR = __builtin_amdgcn_wmma_f32_16x16x4_f32(false, a, false, br, (short)0, accR, false, false);
        accI = __builtin_amdgcn_wmma_f32_16x16x4_f32(false, a, false, bi, (short)0, accI, false, false);
    }

    const int bm = bm0 + laneLo;
    const int b  = bm >> 8;                       // batch
    const int n  = bm & (MODES_ - 1);             // spatial index (< 256)
    #pragma unroll
    for (int v = 0; v < 8; ++v) {
        const int k2 = kb + v + 8 * laneHi;
        v2f z;
        z.x = gelu_exact(accR[v] + lin_b[k2]);    // real gets lin_b
        z.y = gelu_exact(accI[v]);                // imag: no lin_b
        const size_t idx = (((size_t)b * H_ + k2) * NC_ + n) * 2;
        __builtin_nontemporal_store(z, reinterpret_cast<v2f*>(out + idx));
    }
}

// ---------------------------------------------------------------------------
// Kernel 3: tail (n >= MODES). yr == bias (batch-independent), yi == 0, so
//   zr[b,k2,n] = gelu(sum_h lin_w[k2,h]*bias[h,n] + lin_b[k2])  (same all b)
//   zi[b,k2,n] = 0
// One wave = one 16(k2) x 16(n) WMMA tile, broadcast-stored to 4 batches.
// grid: (NC-MODES)/16 = 4080 blocks of 256 threads (8 waves cover k2=0..127).
// ---------------------------------------------------------------------------
__global__ void __launch_bounds__(256)
sfno_bias_tail_kernel(const float* __restrict__ bias,
                      const float* __restrict__ lin_w,
                      const float* __restrict__ lin_b,
                      float* __restrict__ out)
{
    const int lane   = threadIdx.x & 31;
    const int wave   = threadIdx.x >> 5;
    const int laneLo = lane & 15;
    const int laneHi = lane >> 4;
    const int kb     = wave << 4;
    const int n0     = MODES_ + (blockIdx.x << 4);

    v8f acc = {};
    for (int h = 0; h < H_; h += 4) {
        v2f a, b;
        // A = lin_w[k2][h]  (16x4)
        const float* ap = lin_w + (kb + laneLo) * H_ + h + 2 * laneHi;
        a.x = ap[0];  a.y = ap[1];
        // B = bias[h][n]    (4x16), rows are NC_ floats apart
        const float* bp = bias + (size_t)(h + 2 * laneHi) * NC_ + n0 + laneLo;
        b.x = bp[0];  b.y = bp[NC_];
        acc = __builtin_amdgcn_wmma_f32_16x16x4_f32(false, a, false, b, (short)0, acc, false, false);
    }

    const int n = n0 + laneLo;
    #pragma unroll
    for (int v = 0; v < 8; ++v) {
        const int k2 = kb + v + 8 * laneHi;
        v2f z;
        z.x = gelu_exact(acc[v] + lin_b[k2]);
        z.y = 0.0f;                                // gelu(0) == 0
        #pragma unroll
        for (int b4 = 0; b4 < B_; ++b4) {          // broadcast over batch
            const size_t idx = (((size_t)b4 * H_ + k2) * NC_ + n) * 2;
            __builtin_nontemporal_store(z, reinterpret_cast<v2f*>(out + idx));
        }
    }
}

extern "C" void kernel_launch(void* const* d_in, const int* in_sizes, int n_in,
                              void* d_out, int out_size, void* d_ws, size_t ws_size,
                              hipStream_t stream) {
    const float* x      = (const float*)d_in[0];
    const float* w_real = (const float*)d_in[1];
    const float* w_imag = (const float*)d_in[2];
    const float* bias   = (const float*)d_in[3];
    const float* lin_w  = (const float*)d_in[4];
    const float* lin_b  = (const float*)d_in[5];
    float* out = (float*)d_out;

    // workspace: Yr, Yi each H_ x (B_*MODES_) fp32  (1 MB total)
    float* Yr = (float*)d_ws;
    float* Yi = Yr + (size_t)H_ * B_ * MODES_;

    sfno_spectral_kernel<<<dim3(MODES_ / 32), dim3(256), 0, stream>>>(
        x, w_real, w_imag, bias, Yr, Yi);
    sfno_head_kernel<<<dim3((B_ * MODES_) / 16), dim3(256), 0, stream>>>(
        Yr, Yi, lin_w, lin_b, out);
    sfno_bias_tail_kernel<<<dim3((NC_ - MODES_) / 16), dim3(256), 0, stream>>>(
        bias, lin_w, lin_b, out);
}